// GraphConvolutionBase_3023656976523
// MI455X (gfx1250) — compile-verified
//
#include <hip/hip_runtime.h>

typedef __attribute__((ext_vector_type(2))) float v2f;
typedef __attribute__((ext_vector_type(8))) float v8f;

#define D 128  // D_IN == D_OUT == 128

// ---------------------------------------------------------------------------
// Kernel 1: dual GEMM via V_WMMA_F32_16X16X4_F32 (full f32 precision)
//   y   = x @ W      (to workspace; feeds the edge scatter)
//   out = x @ W_r    (residual term; scatter accumulates on top of it)
// Block = 8 waves; block b owns rows [16b, 16b+16); wave w owns cols [16w, 16w+16).
// EXEC is all-ones throughout (no divergence) as WMMA requires.
// ---------------------------------------------------------------------------
__global__ __launch_bounds__(256)
void gcn_dual_gemm_wmma(const float* __restrict__ x,
                        const float* __restrict__ W,
                        const float* __restrict__ Wr,
                        float* __restrict__ y,
                        float* __restrict__ out)
{
    const int wave = threadIdx.x >> 5;          // 0..7  -> column tile
    const int lane = threadIdx.x & 31;
    const int row_base = blockIdx.x * 16;
    const int col_base = wave * 16;

    const int r  = lane & 15;                   // matrix row (A) / col (B,C,D)
    const int hi = lane >> 4;                   // lane-half selects K pair / row half

    const float* arow = x  + (size_t)(row_base + r) * D;
    const float* bW   = W  + col_base + r;
    const float* bWr  = Wr + col_base + r;

    v8f accW = {};   // accumulates x @ W
    v8f accR = {};   // accumulates x @ W_r

    #pragma unroll 4
    for (int kk = 0; kk < D / 4; ++kk) {
        const int kb = kk * 4 + hi * 2;         // this lane's K pair base

        // A fragment (16x4 f32): contiguous pair from this lane's row
        v2f a;
        a.x = arow[kb + 0];
        a.y = arow[kb + 1];

        // B fragments (4x16 f32): rows kb, kb+1 of each weight, column r
        v2f b0, b1;
        b0.x = bW [(size_t)(kb + 0) * D];
        b0.y = bW [(size_t)(kb + 1) * D];
        b1.x = bWr[(size_t)(kb + 0) * D];
        b1.y = bWr[(size_t)(kb + 1) * D];

        accW = __builtin_amdgcn_wmma_f32_16x16x4_f32(
            false, a, false, b0, (short)0, accW, false, false);
        accR = __builtin_amdgcn_wmma_f32_16x16x4_f32(
            false, a, false, b1, (short)0, accR, false, false);
    }

    // C/D layout: VGPR i holds row (row_base + i + 8*hi), col = col_base + r
    const int col = col_base + r;
    #pragma unroll
    for (int i = 0; i < 8; ++i) {
        const size_t row = (size_t)(row_base + i + 8 * hi);
        y  [row * D + col] = accW[i];
        out[row * D + col] = accR[i];
    }
}

// ---------------------------------------------------------------------------
// Kernel 2: edge scatter  out[dst] += val * y[src]
// One wave32 per edge: lane handles a float4 chunk (32 lanes * 4 = 128 feats).
// src/dst/val loads are wave-uniform (broadcast); y reads hit L2 (y = 51 MB
// fits in the 192 MB L2 with avg degree 16 reuse); 4 coalesced f32 atomics.
// ---------------------------------------------------------------------------
__global__ __launch_bounds__(256)
void gcn_edge_scatter(const float* __restrict__ y,
                      const float* __restrict__ adj_val,
                      const int*   __restrict__ adj_src,
                      const int*   __restrict__ adj_dst,
                      float* __restrict__ out,
                      int n_edges)
{
    const int gid = blockIdx.x * blockDim.x + threadIdx.x;
    const int e = gid >> 5;
    if (e >= n_edges) return;

    const int chunk = (gid & 31) * 4;
    const int   s = adj_src[e];
    const int   d = adj_dst[e];
    const float v = adj_val[e];

    const float4 yv = *reinterpret_cast<const float4*>(y + (size_t)s * D + chunk);
    float* o = out + (size_t)d * D + chunk;

    atomicAdd(o + 0, v * yv.x);
    atomicAdd(o + 1, v * yv.y);
    atomicAdd(o + 2, v * yv.z);
    atomicAdd(o + 3, v * yv.w);
}

// ---------------------------------------------------------------------------
// Inputs (setup_inputs order):
//   0: x [N*128 f32]  1: h0 (unused)  2: adj_val [E f32]
//   3: weight [128*128 f32]  4: weight_r [128*128 f32]
//   5: adj_src [E i32]  6: adj_dst [E i32]
// Output: out [N*128 f32]. Workspace: y = x@W, N*128*4 = 51.2 MB.
// ---------------------------------------------------------------------------
extern "C" void kernel_launch(void* const* d_in, const int* in_sizes, int n_in,
                              void* d_out, int out_size, void* d_ws, size_t ws_size,
                              hipStream_t stream)
{
    const float* x       = (const float*)d_in[0];
    const float* adj_val = (const float*)d_in[2];
    const float* W       = (const float*)d_in[3];
    const float* Wr      = (const float*)d_in[4];
    const int*   adj_src = (const int*)d_in[5];
    const int*   adj_dst = (const int*)d_in[6];
    float* out = (float*)d_out;
    float* y   = (float*)d_ws;                  // N*D floats of scratch

    const int n_nodes = in_sizes[0] / D;        // 100000
    const int n_edges = in_sizes[2];            // 1600000

    // 1) out = x@Wr, y = x@W   (16-row stripes; 100000 is divisible by 16)
    gcn_dual_gemm_wmma<<<n_nodes / 16, 256, 0, stream>>>(x, W, Wr, y, out);

    // 2) out[dst] += val * y[src]  (32 lanes per edge)
    const long long work = (long long)n_edges * 32;
    const int blocks = (int)((work + 255) / 256);
    gcn_edge_scatter<<<blocks, 256, 0, stream>>>(y, adj_val, adj_src, adj_dst,
                                                 out, n_edges);
}